// ExternalInteraction_65609920413984
// MI455X (gfx1250) — compile-verified
//
#include <hip/hip_runtime.h>

typedef __attribute__((ext_vector_type(2))) float v2f;
typedef __attribute__((ext_vector_type(8))) float v8f;

#define BB 32
#define UU 128
#define II 256
#define DD 256

// Pass 1: per-batch column sums via V_WMMA_F32_16X16X4_F32 with an all-ones
// A matrix.  C += ones(16x4) x B(4x16) accumulated over K = rows gives every
// row of C equal to the column sums of the 16-wide d-tile.
// sums layout in workspace: [0, BB*DD) = img_sum, [BB*DD, 2*BB*DD) = user_sum.
__global__ __launch_bounds__(32) void colsum_wmma_kernel(
    const float* __restrict__ user, const float* __restrict__ img,
    float* __restrict__ sums)
{
    const int task = blockIdx.x;      // BB * 32 tasks, one wave each
    const int b    = task >> 5;
    const int t    = task & 31;       // 0..15 -> img tile, 16..31 -> user tile

    const float* src;
    int rows;
    float* out;
    if (t < 16) {
        src  = img + (size_t)b * II * DD;
        rows = II;
        out  = sums + b * DD + t * 16;
    } else {
        src  = user + (size_t)b * UU * DD;
        rows = UU;
        out  = sums + BB * DD + b * DD + (t - 16) * 16;
    }

    const int lane  = threadIdx.x;    // wave32
    const int n     = lane & 15;      // column within the 16-wide d tile
    const int khalf = lane >> 4;      // which K pair this half-wave supplies
    const int d0    = (t & 15) * 16;

    // lane walks its column; rows k0+2*khalf and k0+1+2*khalf feed B VGPR0/1
    const float* p = src + d0 + n + (size_t)(2 * khalf) * DD;

    v2f a;
    a.x = 1.0f;
    a.y = 1.0f;
    v8f c = {};

    for (int k0 = 0; k0 < rows; k0 += 4) {
        v2f bm;
        bm.x = p[0];
        bm.y = p[DD];
        __builtin_prefetch(p + 4 * DD, 0, 3);   // global_prefetch_b8, next block of rows
        p += 4 * DD;
        // 8 args: (neg_a, A, neg_b, B, c_mod, C, reuse_a, reuse_b)
        c = __builtin_amdgcn_wmma_f32_16x16x4_f32(
                /*neg_a=*/false, a, /*neg_b=*/false, bm,
                /*c_mod=*/(short)0, c, /*reuse_a=*/false, /*reuse_b=*/false);
    }

    // C row M=0 lives in VGPR0 on lanes 0..15 (N = lane); all rows are equal
    // anyway since A is all-ones.
    if (lane < 16) out[n] = c[0];
}

// Pass 2: out[b,r,d] = in[b,r,d] * sums[b,d], vectorized as float4 (b128).
// perBatchShift = log2(rows * DD / 4); power of two for both tensors.
__global__ __launch_bounds__(256) void scale_kernel(
    const float4* __restrict__ in, const float* __restrict__ sums,
    float4* __restrict__ out, int perBatchShift)
{
    const int idx = blockIdx.x * blockDim.x + threadIdx.x;  // float4 index
    const int b   = idx >> perBatchShift;
    const int d4  = idx & (DD / 4 - 1);

    const float4 s = ((const float4*)(sums + b * DD))[d4];
    float4 v = in[idx];
    v.x *= s.x;
    v.y *= s.y;
    v.z *= s.z;
    v.w *= s.w;
    out[idx] = v;
}

extern "C" void kernel_launch(void* const* d_in, const int* in_sizes, int n_in,
                              void* d_out, int out_size, void* d_ws, size_t ws_size,
                              hipStream_t stream) {
    const float* user = (const float*)d_in[0];   // [32,128,256] fp32
    const float* img  = (const float*)d_in[1];   // [32,256,256] fp32
    float* out  = (float*)d_out;                 // user-out ++ img-out
    float* sums = (float*)d_ws;                  // 2 * 32 * 256 floats = 64 KB

    // Pass 1: 32 batches x (16 img tiles + 16 user tiles), one wave32 each.
    colsum_wmma_kernel<<<BB * 32, 32, 0, stream>>>(user, img, sums);

    const int userTotal4 = BB * UU * DD / 4;     // 262144 float4
    const int imgTotal4  = BB * II * DD / 4;     // 524288 float4

    // out_user[b,u,d] = user * img_sum   (img_sum at sums + 0)
    scale_kernel<<<userTotal4 / 256, 256, 0, stream>>>(
        (const float4*)user, sums, (float4*)out, /*log2(128*64)=*/13);

    // out_img[b,i,d] = img * user_sum    (user_sum at sums + BB*DD)
    scale_kernel<<<imgTotal4 / 256, 256, 0, stream>>>(
        (const float4*)img, sums + BB * DD, (float4*)(out + BB * UU * DD),
        /*log2(256*64)=*/14);
}